// PointcloudSubsampling_62603443306512
// MI455X (gfx1250) — compile-verified
//
#include <hip/hip_runtime.h>
#include <stdint.h>

// Problem constants (fixed by the reference).
#define BATCH 16
#define NPTS  100000
#define CCH   16
#define KSEL  1024
#define TILE  1024                       // points per block-wide tile (1 per thread)
#define NT    ((NPTS + TILE - 1) / TILE) // 98 tiles
#define DEPTH 8                          // async staging slots per wave (8-deep pipeline)

// ---------------------------------------------------------------------------
// Kernel 1: pack xyz -> float4 (16B aligned for async b128) and reset min_d.
// Runs every launch so the workspace state is re-initialized deterministically.
// ---------------------------------------------------------------------------
__global__ __launch_bounds__(256) void fps_pack(const float* __restrict__ points,
                                                float4* __restrict__ xyz4,
                                                float* __restrict__ mind) {
    size_t i = (size_t)blockIdx.x * blockDim.x + threadIdx.x;
    if (i >= (size_t)BATCH * NPTS) return;
    const float* p = points + i * CCH;
    xyz4[i] = make_float4(p[0], p[1], p[2], 0.0f);
    mind[i] = __builtin_inff();
}

// Issue one per-wave async tile load (32 lanes x 16B) into an LDS slot.
// s_wait_dscnt 0x0 first: guarantees any pending ds_load of the slot being
// recycled has fully retired before the async engine overwrites it.
#define ASYNC_ISSUE(ldsaddr, voff, sbase)                                     \
    asm volatile("s_wait_dscnt 0x0\n\t"                                       \
                 "global_load_async_to_lds_b128 %0, %1, %2"                   \
                 :: "v"(ldsaddr), "v"(voff), "s"(sbase) : "memory")

// ---------------------------------------------------------------------------
// Kernel 2: farthest point sampling, one block (32 wave32 waves) per batch.
// xyz tiles stream through 8-deep per-wave LDS slots via CDNA5 async
// global->LDS loads (ASYNCcnt pipelining; no staging barriers since each wave
// only reads LDS its own async loads produced). min_d is software-pipelined
// one tile ahead in a register to hide its L2 load latency.
// ---------------------------------------------------------------------------
__global__ __launch_bounds__(1024, 1) void fps_main(const float* __restrict__ points,
                                                    const int* __restrict__ start_idx,
                                                    float* __restrict__ out,
                                                    const float4* __restrict__ xyz4,
                                                    float* __restrict__ mind) {
    __shared__ float4 stage[32][DEPTH][32];   // [wave][slot][lane] = 128 KB
    __shared__ float  s_val[32];
    __shared__ int    s_idx[32];
    __shared__ int    s_cur;

    const int b    = blockIdx.x;
    const int tid  = threadIdx.x;
    const int lane = tid & 31;
    const int w    = tid >> 5;

    const float4* xb = xyz4 + (size_t)b * NPTS;
    float*        mb = mind + (size_t)b * NPTS;
    const float*  pb = points + (size_t)b * NPTS * CCH;
    float*        ob = out + (size_t)b * KSEL * CCH;

    // Uniform SGPR base for async loads (batch slice of packed xyz).
    const unsigned long long sbase = (unsigned long long)(uintptr_t)xb;
    // Per-lane LDS byte offset of slot 0 (low 32 bits of the generic address
    // of a __shared__ object == DS-relative offset); slots are +512B apart.
    const uint32_t ldsS0 = (uint32_t)(uintptr_t)(&stage[w][0][lane]);

    int cur = __builtin_amdgcn_readfirstlane(start_idx[b]);

    for (int k = 0; k < KSEL; ++k) {
        // Emit output row k = all 16 channels of the currently selected point.
        if (tid < CCH) ob[k * CCH + tid] = pb[(size_t)cur * CCH + tid];

        // Selected point coords (uniform/scalar address).
        const float4 sel = xb[cur];

        float bv = -__builtin_inff();
        int   bi = 0x7fffffff;

        // Prologue: fill pipeline with tiles 0..DEPTH-2 (NT >> DEPTH).
        #pragma unroll
        for (int pt = 0; pt < DEPTH - 1; ++pt) {
            uint32_t voff = (uint32_t)(pt * TILE + tid) * 16u;
            uint32_t ldsn = ldsS0 + (uint32_t)pt * (32u * (uint32_t)sizeof(float4));
            ASYNC_ISSUE(ldsn, voff, sbase);
        }
        // Pipelined min_d for tile 0.
        float md = mb[tid];

        for (int t = 0; t < NT; ++t) {
            if (t + DEPTH - 1 < NT) {
                // Top up the pipeline: tile t+7 -> slot (t-1)%8 (read last iter).
                int in = (t + DEPTH - 1) * TILE + tid;
                if (in >= NPTS) in = NPTS - 1;
                uint32_t voff = (uint32_t)in * 16u;
                uint32_t ldsn = ldsS0 +
                    (uint32_t)((t + DEPTH - 1) & (DEPTH - 1)) * (32u * (uint32_t)sizeof(float4));
                ASYNC_ISSUE(ldsn, voff, sbase);
                // 8 in flight; retire the oldest (tile t).
                asm volatile("s_wait_asynccnt 0x7" ::: "memory");
            } else {
                asm volatile("s_wait_asynccnt 0x0" ::: "memory");
            }

            const float4 p = stage[w][t & (DEPTH - 1)][lane];
            const int i = t * TILE + tid;

            // Prefetch next tile's min_d (clamped tail read is benign/unused).
            int inm = i + TILE;
            if (inm >= NPTS) inm = NPTS - 1;
            const float md_next = mb[inm];

            if (i < NPTS) {
                const float dx = p.x - sel.x;
                const float dy = p.y - sel.y;
                const float dz = p.z - sel.z;
                const float d  = dx * dx + dy * dy + dz * dz;
                const float nm = fminf(md, d);      // per-thread-private RMW
                mb[i] = nm;
                if (nm > bv || (nm == bv && i < bi)) { bv = nm; bi = i; }
            }
            md = md_next;
        }

        // Wave32 argmax reduction, first-index tie-break (matches jnp.argmax).
        #pragma unroll
        for (int off = 16; off > 0; off >>= 1) {
            float ov = __shfl_xor(bv, off, 32);
            int   oi = __shfl_xor(bi, off, 32);
            if (ov > bv || (ov == bv && oi < bi)) { bv = ov; bi = oi; }
        }
        if (lane == 0) { s_val[w] = bv; s_idx[w] = bi; }
        __syncthreads();
        if (w == 0) {
            bv = s_val[lane];
            bi = s_idx[lane];
            #pragma unroll
            for (int off = 16; off > 0; off >>= 1) {
                float ov = __shfl_xor(bv, off, 32);
                int   oi = __shfl_xor(bi, off, 32);
                if (ov > bv || (ov == bv && oi < bi)) { bv = ov; bi = oi; }
            }
            if (lane == 0) s_cur = bi;
        }
        __syncthreads();
        cur = __builtin_amdgcn_readfirstlane(s_cur);
    }
}

// ---------------------------------------------------------------------------
// Launch: d_in[0] = points [16,100000,16] f32, d_in[1] = start_idx [16] i32.
// d_out = [16,1024,16] f32. Workspace: 25.6 MB packed xyz4 + 6.4 MB min_d.
// ---------------------------------------------------------------------------
extern "C" void kernel_launch(void* const* d_in, const int* in_sizes, int n_in,
                              void* d_out, int out_size, void* d_ws, size_t ws_size,
                              hipStream_t stream) {
    (void)in_sizes; (void)n_in; (void)out_size; (void)ws_size;

    const float* points = (const float*)d_in[0];
    const int*   start  = (const int*)d_in[1];
    float*       out    = (float*)d_out;

    float4* xyz4 = (float4*)d_ws;
    float*  mind = (float*)((char*)d_ws + (size_t)BATCH * NPTS * sizeof(float4));

    const int total = BATCH * NPTS;
    fps_pack<<<(total + 255) / 256, 256, 0, stream>>>(points, xyz4, mind);
    fps_main<<<BATCH, 1024, 0, stream>>>(points, start, out, xyz4, mind);
}